// GliZNetRepresentationAggregator_19112604467777
// MI455X (gfx1250) — compile-verified
//
#include <hip/hip_runtime.h>

typedef __attribute__((ext_vector_type(2))) float v2f;
typedef __attribute__((ext_vector_type(8))) float v8f;

#define B_   8
#define S_   4096
#define H_   1024
#define P_   1024
#define L_   64
#define EPS_ 1e-8f

// ---------------------------------------------------------------------------
// Workspace layout (floats):
//   cls : [0,      8192)   cls_tokens[b,p]      (B x P)
//   v   : [8192,  16384)   v[b,h] = W_lab @ cls (B x H)
//   c0  : [16384, 16392)   c0[b]  = b_lab . cls
//   num : [16392, 16904)   scatter numerators   (B*L)
//   den : [16904, 17416)   scatter weight sums  (B*L)
// ---------------------------------------------------------------------------

__global__ __launch_bounds__(512)
void k_init(float* __restrict__ num, float* __restrict__ den,
            float* __restrict__ c0) {
  int i = blockIdx.x * blockDim.x + threadIdx.x;
  if (i < B_ * L_) { num[i] = 0.0f; den[i] = 0.0f; }
  if (i < B_) c0[i] = 0.0f;
}

// cls[b, col] = hidden_states[b, 0, :] . W_cls[:, col] + b_cls[col]
// One wave (32 lanes) per 16-column tile; K-loop of v_wmma_f32_16x16x4_f32.
// A-matrix 16x4 f32 layout: lanes 0-15 hold M=0..15 with K=0,1 in a.x,a.y;
// lanes 16-31 hold the same rows with K=2,3. Rows >= 8 are zero-masked.
__global__ __launch_bounds__(32)
void k_cls(const float* __restrict__ hs, const float* __restrict__ Wc,
           const float* __restrict__ bc, float* __restrict__ cls) {
  const int tile = blockIdx.x;           // 64 tiles of 16 columns
  const int lane = threadIdx.x;
  const int half = lane >> 4;            // K sub-block (0 -> K+0/1, 1 -> K+2/3)
  const int m    = lane & 15;
  const int col  = tile * 16 + m;
  const float amask = (m < B_) ? 1.0f : 0.0f;
  const size_t abase = (size_t)((m < B_) ? m : 0) * S_ * H_;  // row b, s=0

  v8f acc = {};
  for (int k = 0; k < H_; k += 4) {
    const int kk = k + half * 2;
    v2f a, b;
    a.x = hs[abase + kk]     * amask;
    a.y = hs[abase + kk + 1] * amask;
    b.x = Wc[(size_t)kk * P_ + col];
    b.y = Wc[(size_t)(kk + 1) * P_ + col];
    acc = __builtin_amdgcn_wmma_f32_16x16x4_f32(false, a, false, b,
                                                (short)0, acc, false, false);
  }
  // D layout: VGPR j holds M=j on lanes 0-15 (M=j+8 on lanes 16-31, unused).
  if (lane < 16) {
    const float bias = bc[col];
#pragma unroll
    for (int j = 0; j < B_; ++j)
      cls[(size_t)j * P_ + col] = acc[j] + bias;
  }
}

// v[b, n] = sum_p cls[b, p] * W_lab[n, p]   (B = W_lab read transposed)
__global__ __launch_bounds__(32)
void k_v(const float* __restrict__ cls, const float* __restrict__ Wl,
         float* __restrict__ v) {
  const int tile = blockIdx.x;           // 64 tiles of 16 output rows of H
  const int lane = threadIdx.x;
  const int half = lane >> 4;
  const int m    = lane & 15;
  const int n    = tile * 16 + m;        // output index in H
  const float amask = (m < B_) ? 1.0f : 0.0f;
  const size_t arow = (size_t)((m < B_) ? m : 0) * P_;

  v8f acc = {};
  for (int k = 0; k < P_; k += 4) {
    const int kk = k + half * 2;
    v2f a, b;
    a.x = cls[arow + kk]     * amask;
    a.y = cls[arow + kk + 1] * amask;
    b.x = Wl[(size_t)n * P_ + kk];       // W_lab[n, kk]
    b.y = Wl[(size_t)n * P_ + kk + 1];
    acc = __builtin_amdgcn_wmma_f32_16x16x4_f32(false, a, false, b,
                                                (short)0, acc, false, false);
  }
  if (lane < 16) {
#pragma unroll
    for (int j = 0; j < B_; ++j)
      v[(size_t)j * H_ + n] = acc[j];
  }
}

// c0[b] = b_lab . cls[b]   (one wave32 per batch)
__global__ __launch_bounds__(256)
void k_c0(const float* __restrict__ cls, const float* __restrict__ bl,
          float* __restrict__ c0) {
  const int b    = threadIdx.x >> 5;
  const int lane = threadIdx.x & 31;
  float s = 0.0f;
  for (int i = lane; i < P_; i += 32)
    s += bl[i] * cls[(size_t)b * P_ + i];
#pragma unroll
  for (int off = 16; off > 0; off >>= 1)
    s += __shfl_xor(s, off, 32);
  if (lane == 0) c0[b] = s;
}

// Dominant pass: d[b,s] = h[b,s] . v[b]; scatter w*(d + c0[b]) into num/den.
// 1 wave per token, 8 tokens (same batch) per 256-thread block; v[b] in LDS.
__global__ __launch_bounds__(256)
void k_dot(const float* __restrict__ hs, const float* __restrict__ vvec,
           const float* __restrict__ c0, const float* __restrict__ caw,
           const int* __restrict__ lm, float* __restrict__ num,
           float* __restrict__ den) {
  __shared__ float vs[H_];
  const int blk = blockIdx.x;                    // 4096 blocks
  const int b   = blk / (S_ / 8);
  const int s0  = (blk % (S_ / 8)) * 8;

  {  // stage v[b] (4 KB) into LDS once per block
    const float4* src = (const float4*)(vvec + (size_t)b * H_);
    float4* dst = (float4*)vs;
    for (int i = threadIdx.x; i < H_ / 4; i += 256) dst[i] = src[i];
  }
  __syncthreads();

  const int wave = threadIdx.x >> 5;
  const int lane = threadIdx.x & 31;
  const int s    = s0 + wave;
  const size_t tok = (size_t)b * S_ + s;

  const float4* hp = (const float4*)(hs + tok * H_);
  const float4* vp = (const float4*)vs;
  float d = 0.0f;
#pragma unroll
  for (int i = 0; i < H_ / 128; ++i) {           // 8 x b128 loads per lane
    const float4 hv = hp[i * 32 + lane];
    const float4 vv = vp[i * 32 + lane];
    d += hv.x * vv.x + hv.y * vv.y + hv.z * vv.z + hv.w * vv.w;
  }
#pragma unroll
  for (int off = 16; off > 0; off >>= 1)
    d += __shfl_xor(d, off, 32);

  if (lane == 0) {
    const int label = lm[tok];
    if (label > 0) {
      const float w = caw[tok];
      const int slot = b * L_ + (label - 1);
      atomicAdd(&num[slot], w * (d + c0[b]));
      atomicAdd(&den[slot], w);
    }
  }
}

// logits = num / max(den, eps); also emit batch indices and label ids.
__global__ __launch_bounds__(256)
void k_final(const float* __restrict__ num, const float* __restrict__ den,
             float* __restrict__ out, int out_size) {
  const int i = blockIdx.x * blockDim.x + threadIdx.x;
  if (i >= B_ * L_) return;
  out[i] = num[i] / fmaxf(den[i], EPS_);
  if (out_size >= 3 * B_ * L_) {
    out[B_ * L_ + i]     = (float)(i / L_);       // final_batch_indices
    out[2 * B_ * L_ + i] = (float)(i % L_ + 1);   // final_label_ids
  }
}

extern "C" void kernel_launch(void* const* d_in, const int* in_sizes, int n_in,
                              void* d_out, int out_size, void* d_ws, size_t ws_size,
                              hipStream_t stream) {
  (void)in_sizes; (void)n_in; (void)ws_size;

  const float* hs  = (const float*)d_in[0];   // [B,S,H]
  const float* caw = (const float*)d_in[1];   // [B,S]
  const float* Wc  = (const float*)d_in[2];   // [H,P]
  const float* bc  = (const float*)d_in[3];   // [P]
  const float* Wl  = (const float*)d_in[4];   // [H,P]
  const float* bl  = (const float*)d_in[5];   // [P]
  const int*   lm  = (const int*)d_in[6];     // [B,S]
  float* out = (float*)d_out;

  float* ws  = (float*)d_ws;
  float* cls = ws;                 // 8192
  float* v   = ws + 8192;          // 8192
  float* c0  = ws + 16384;         // 8
  float* num = ws + 16392;         // 512
  float* den = ws + 16904;         // 512

  k_init <<<1, 512, 0, stream>>>(num, den, c0);
  k_cls  <<<P_ / 16, 32, 0, stream>>>(hs, Wc, bc, cls);
  k_v    <<<H_ / 16, 32, 0, stream>>>(cls, Wl, v);
  k_c0   <<<1, 256, 0, stream>>>(cls, bl, c0);
  k_dot  <<<(B_ * S_) / 8, 256, 0, stream>>>(hs, v, c0, caw, lm, num, den);
  k_final<<<(B_ * L_ + 255) / 256, 256, 0, stream>>>(num, den, out, out_size);
}